// InfrastructureGNN_70978629533877
// MI455X (gfx1250) — compile-verified
//
#include <hip/hip_runtime.h>
#include <math.h>

// ---- problem constants (match reference) ----
#define NN      100000
#define EE      1600000
#define DIN_    24
#define H_      128
#define DOUT_   12
#define NHEADS_ 4
#define DH_     32
#define BN_EPS  1e-5f

typedef __attribute__((ext_vector_type(2))) float v2f;
typedef __attribute__((ext_vector_type(8))) float v8f;

__device__ __forceinline__ float lrelu02(float v) { return v > 0.f ? v : 0.2f * v; }

// ordered-int trick: correct for mixed-sign IEEE floats
__device__ __forceinline__ void atomicMaxF(float* addr, float val) {
  if (val >= 0.f) atomicMax((int*)addr, __float_as_int(val));
  else            atomicMin((unsigned int*)addr, __float_as_uint(val));
}

// ---------------------------------------------------------------------------
// Weight pre-transpose with zero-padding: Bt[Ppad x K], Bt[n*K+k] = B[k*P+n]
// (rows n >= P are zero so WMMA B-fragment loads need no bounds guard)
// ---------------------------------------------------------------------------
__global__ void k_transpose(const float* __restrict__ B, float* __restrict__ Bt,
                            int K, int P, int Ppad) {
  int idx = blockIdx.x * blockDim.x + threadIdx.x;
  if (idx >= Ppad * K) return;
  int n = idx / K, k = idx - n * K;
  Bt[idx] = (n < P) ? B[k * P + n] : 0.f;
}

// ---------------------------------------------------------------------------
// WMMA fp32 GEMM: C[NN x P] = A[NN x K] @ B[K x P] (+bias)(relu?)
// One wave32 = one 16x16 tile via V_WMMA_F32_16X16X4_F32 (fp32 in/accum:
// compute is ~10 GFLOP total and the app is scatter-bound, so full precision
// on the matrix path is free).
// A frag (16x4, ISA layout): lanes 0-15 {K=k,k+1}, lanes 16-31 {K=k+2,k+3}
// B frag (4x16): lane = column, same K packing -> transposed B gives b64 loads
// D: VGPR r -> row mt*16 + r + 8*(lane>=16), col nt*16 + (lane&15)
// ---------------------------------------------------------------------------
template<int K, int P>
__global__ __launch_bounds__(256)
void k_gemm_wmma(const float* __restrict__ A, const float* __restrict__ Bt,
                 const float* __restrict__ bias, float* __restrict__ C, int relu)
{
  constexpr int PT  = (P + 15) / 16;
  constexpr int TOT = (NN / 16) * PT;
  int wave = threadIdx.x >> 5;
  int lane = threadIdx.x & 31;
  int tile = blockIdx.x * 8 + wave;
  if (tile >= TOT) return;                 // wave-uniform; EXEC all-ones at wmma
  int mt = tile / PT;
  int nt = tile - mt * PT;
  int mr = lane & 15;
  int hi = lane >> 4;                      // 0 or 1
  const float* arow = A  + (size_t)(mt * 16 + mr) * K + 2 * hi;
  const float* bcol = Bt + (size_t)(nt * 16 + mr) * K + 2 * hi;

  v8f acc = {};
#pragma unroll
  for (int k = 0; k < K; k += 4) {
    v2f a = *(const v2f*)(arow + k);       // global_load_b64, 8B-aligned
    v2f b = *(const v2f*)(bcol + k);       // global_load_b64, 8B-aligned
    acc = __builtin_amdgcn_wmma_f32_16x16x4_f32(false, a, false, b,
                                                (short)0, acc, false, false);
  }
  int n = nt * 16 + mr;
  if ((P & 15) == 0 || n < P) {
    float bv = bias ? bias[n] : 0.f;
#pragma unroll
    for (int r = 0; r < 8; ++r) {
      float v = acc[r] + bv;
      if (relu) v = fmaxf(v, 0.f);
      C[(size_t)(mt * 16 + r + hi * 8) * P + n] = v;
    }
  }
}

// ---------------- degree / normalization ----------------
__global__ void k_fill1(float* p, int n) {
  int i = blockIdx.x * blockDim.x + threadIdx.x;
  if (i < n) p[i] = 1.0f;                       // self-loop weight
}
__global__ void k_deg_edges(float* deg, const int* __restrict__ ei,
                            const float* __restrict__ w, int e) {
  int i = blockIdx.x * blockDim.x + threadIdx.x;
  if (i < e) atomicAdd(&deg[ei[EE + i]], w[i]);
}
__global__ void k_dinv(float* d, int n) {
  int i = blockIdx.x * blockDim.x + threadIdx.x;
  if (i < n) { float v = d[i]; d[i] = v > 0.f ? rsqrtf(v) : 0.f; }
}

// ---------------- GCN aggregation, H=128 vectorized float4 ----------------
__global__ void k_gcn_self128v(const float* __restrict__ h, const float* __restrict__ dinv,
                               float* __restrict__ out) {
  int idx = blockIdx.x * blockDim.x + threadIdx.x;          // NN*32
  if (idx >= NN * 32) return;
  int i = idx >> 5;
  float di = dinv[i], c = di * di;
  float4 hv = ((const float4*)h)[idx];
  ((float4*)out)[idx] = make_float4(hv.x * c, hv.y * c, hv.z * c, hv.w * c);
}
__global__ void k_gcn_edges128v(const float* __restrict__ h, const float* __restrict__ dinv,
                                const float* __restrict__ w, const int* __restrict__ ei,
                                float* __restrict__ out) {
  int idx = blockIdx.x * blockDim.x + threadIdx.x;          // EE*32
  if (idx >= EE * 32) return;
  int e = idx >> 5, q = idx & 31;
  int s = ei[e], d = ei[EE + e];
  float coef = w[e] * dinv[s] * dinv[d];
  float4 hv = ((const float4*)(h + (size_t)s * H_))[q];     // b128 gather
  float* op = out + (size_t)d * H_ + q * 4;
  atomicAdd(op + 0, coef * hv.x); atomicAdd(op + 1, coef * hv.y);
  atomicAdd(op + 2, coef * hv.z); atomicAdd(op + 3, coef * hv.w);
}
// C=12 head: scalar
template<int C>
__global__ void k_gcn_self(const float* __restrict__ h, const float* __restrict__ dinv,
                           float* __restrict__ out) {
  int idx = blockIdx.x * blockDim.x + threadIdx.x;
  if (idx >= NN * C) return;
  int i = idx / C;
  float di = dinv[i];
  out[idx] = di * di * h[idx];
}
template<int C>
__global__ void k_gcn_edges(const float* __restrict__ h, const float* __restrict__ dinv,
                            const float* __restrict__ w, const int* __restrict__ ei,
                            float* __restrict__ out) {
  int idx = blockIdx.x * blockDim.x + threadIdx.x;
  if (idx >= EE * C) return;
  int e = idx / C, c = idx - e * C;
  int s = ei[e], d = ei[EE + e];
  float coef = w[e] * dinv[s] * dinv[d];
  atomicAdd(&out[d * C + c], coef * h[s * C + c]);
}

// ---------------- batch-norm (two-pass, scale/shift folded) ----------------
__global__ __launch_bounds__(H_)
void k_bn_stats(const float* __restrict__ h, const float* __restrict__ bias,
                float* __restrict__ stats) {
  int c = threadIdx.x;
  float b = bias[c], s = 0.f, s2 = 0.f;
  int r0 = blockIdx.x * 256;
  int r1 = r0 + 256; if (r1 > NN) r1 = NN;
  for (int r = r0; r < r1; ++r) {
    float v = h[r * H_ + c] + b;
    s += v; s2 += v * v;
  }
  atomicAdd(&stats[c], s);
  atomicAdd(&stats[H_ + c], s2);
}
__global__ void k_bn_finalize(const float* __restrict__ stats, const float* __restrict__ g,
                              const float* __restrict__ be, float* __restrict__ kAB) {
  int c = threadIdx.x;
  if (c >= H_) return;
  float m  = stats[c] * (1.0f / NN);
  float v  = stats[H_ + c] * (1.0f / NN) - m * m;
  float is = rsqrtf(v + BN_EPS);
  float a  = is * g[c];
  kAB[c] = a;
  kAB[H_ + c] = be[c] - m * a;
}
__global__ void k_bn_apply(const float* __restrict__ h, const float* __restrict__ bias,
                           const float* __restrict__ kAB, const float* __restrict__ res,
                           float* __restrict__ out) {
  int idx = blockIdx.x * blockDim.x + threadIdx.x;          // NN*32
  if (idx >= NN * 32) return;
  int q = idx & 31;
  float4 hv = ((const float4*)h)[idx];
  float4 bv = ((const float4*)bias)[q];
  float4 ka = ((const float4*)kAB)[q];
  float4 kb = ((const float4*)(kAB + H_))[q];
  float4 o;
  o.x = fmaxf((hv.x + bv.x) * ka.x + kb.x, 0.f);
  o.y = fmaxf((hv.y + bv.y) * ka.y + kb.y, 0.f);
  o.z = fmaxf((hv.z + bv.z) * ka.z + kb.z, 0.f);
  o.w = fmaxf((hv.w + bv.w) * ka.w + kb.w, 0.f);
  if (res) {
    float4 rv = ((const float4*)res)[idx];
    o.x += rv.x; o.y += rv.y; o.z += rv.z; o.w += rv.w;
  }
  ((float4*)out)[idx] = o;
}

// ---------------- GAT attention ----------------
__global__ void k_att_scores(const float* __restrict__ hg, const float* __restrict__ asv,
                             const float* __restrict__ adv, float* __restrict__ as_,
                             float* __restrict__ ad_) {
  int idx = blockIdx.x * blockDim.x + threadIdx.x;
  if (idx >= NN * NHEADS_) return;
  int i = idx >> 2, hd = idx & 3;
  const float* hp = hg + (size_t)i * H_ + hd * DH_;
  const float* sp = asv + hd * DH_;
  const float* dp = adv + hd * DH_;
  float vs = 0.f, vd = 0.f;
#pragma unroll
  for (int k = 0; k < DH_; ++k) { float hv = hp[k]; vs += hv * sp[k]; vd += hv * dp[k]; }
  as_[idx] = vs; ad_[idx] = vd;
}
__global__ void k_att_maxinit(const float* __restrict__ as_, const float* __restrict__ ad_,
                              float* __restrict__ emax) {
  int idx = blockIdx.x * blockDim.x + threadIdx.x;
  if (idx >= NN * NHEADS_) return;
  emax[idx] = lrelu02(as_[idx] + ad_[idx]);           // self-loop score
}
__global__ void k_att_max(const int* __restrict__ ei, const float* __restrict__ as_,
                          const float* __restrict__ ad_, float* __restrict__ emax) {
  int idx = blockIdx.x * blockDim.x + threadIdx.x;
  if (idx >= EE * NHEADS_) return;
  int e = idx >> 2, hd = idx & 3;
  int s = ei[e], d = ei[EE + e];
  atomicMaxF(&emax[d * NHEADS_ + hd],
             lrelu02(as_[s * NHEADS_ + hd] + ad_[d * NHEADS_ + hd]));
}
__global__ void k_att_suminit(const float* __restrict__ as_, const float* __restrict__ ad_,
                              const float* __restrict__ emax, float* __restrict__ esum) {
  int idx = blockIdx.x * blockDim.x + threadIdx.x;
  if (idx >= NN * NHEADS_) return;
  esum[idx] = expf(lrelu02(as_[idx] + ad_[idx]) - emax[idx]);   // self term
}
__global__ void k_att_sum(const int* __restrict__ ei, const float* __restrict__ as_,
                          const float* __restrict__ ad_, const float* __restrict__ emax,
                          float* __restrict__ esum, float* __restrict__ exbuf) {
  int idx = blockIdx.x * blockDim.x + threadIdx.x;
  if (idx >= EE * NHEADS_) return;
  int e = idx >> 2, hd = idx & 3;
  int s = ei[e], d = ei[EE + e];
  float ex = expf(lrelu02(as_[s * NHEADS_ + hd] + ad_[d * NHEADS_ + hd])
                  - emax[d * NHEADS_ + hd]);
  exbuf[idx] = ex;                                    // cached for aggregation
  atomicAdd(&esum[d * NHEADS_ + hd], ex);
}
__global__ void k_att_aggselfv(const float* __restrict__ hg, const float* __restrict__ as_,
                               const float* __restrict__ ad_, const float* __restrict__ emax,
                               const float* __restrict__ esum, float* __restrict__ out) {
  int idx = blockIdx.x * blockDim.x + threadIdx.x;    // NN*32
  if (idx >= NN * 32) return;
  int i = idx >> 5, q = idx & 31, hd = q >> 3;
  int ih = i * NHEADS_ + hd;
  float alpha = expf(lrelu02(as_[ih] + ad_[ih]) - emax[ih]) / esum[ih];
  float4 hv = ((const float4*)hg)[idx];
  ((float4*)out)[idx] = make_float4(hv.x * alpha, hv.y * alpha, hv.z * alpha, hv.w * alpha);
}
__global__ void k_att_aggedgesv(const int* __restrict__ ei, const float* __restrict__ hg,
                                const float* __restrict__ exbuf, const float* __restrict__ esum,
                                float* __restrict__ out) {
  int idx = blockIdx.x * blockDim.x + threadIdx.x;    // EE*32
  if (idx >= EE * 32) return;
  int e = idx >> 5, q = idx & 31, hd = q >> 3;
  int s = ei[e], d = ei[EE + e];
  float alpha = exbuf[e * NHEADS_ + hd] / esum[d * NHEADS_ + hd];
  float4 hv = ((const float4*)(hg + (size_t)s * H_))[q];
  float* op = out + (size_t)d * H_ + q * 4;
  atomicAdd(op + 0, alpha * hv.x); atomicAdd(op + 1, alpha * hv.y);
  atomicAdd(op + 2, alpha * hv.z); atomicAdd(op + 3, alpha * hv.w);
}

// ---------------- final fusion ----------------
__global__ void k_final(const float* __restrict__ x, const float* __restrict__ agg4,
                        const float* __restrict__ gout, const float* __restrict__ Ws,
                        const float* __restrict__ bs, const float* __restrict__ b4,
                        float* __restrict__ out) {
  int idx = blockIdx.x * blockDim.x + threadIdx.x;
  if (idx >= NN * DOUT_) return;
  int i = idx / DOUT_, j = idx - i * DOUT_;
  float status = x[i * DIN_ + 12];
  float flag = status < 0.5f ? 1.f : 0.f;
  float gate = 1.f / (1.f + expf(-gout[idx]));
  float sig  = (1.f - status) * Ws[j] + bs[j];
  out[idx] = (agg4[idx] + b4[j]) + 2.5f * flag * gate * sig;
}

// ---------------------------------------------------------------------------
extern "C" void kernel_launch(void* const* d_in, const int* in_sizes, int n_in,
                              void* d_out, int out_size, void* d_ws, size_t ws_size,
                              hipStream_t stream) {
  const float* x    = (const float*)d_in[0];
  const int*   ei   = (const int*)  d_in[1];
  const float* ew   = (const float*)d_in[2];
  const float* W1   = (const float*)d_in[3];
  const float* b1   = (const float*)d_in[4];
  const float* g1   = (const float*)d_in[5];
  const float* be1  = (const float*)d_in[6];
  const float* Wg   = (const float*)d_in[7];
  const float* asv  = (const float*)d_in[8];
  const float* adv  = (const float*)d_in[9];
  const float* bg   = (const float*)d_in[10];
  const float* g2   = (const float*)d_in[11];
  const float* be2  = (const float*)d_in[12];
  const float* W3   = (const float*)d_in[13];
  const float* b3   = (const float*)d_in[14];
  const float* g3   = (const float*)d_in[15];
  const float* be3  = (const float*)d_in[16];
  const float* W4   = (const float*)d_in[17];
  const float* b4   = (const float*)d_in[18];
  const float* Wi   = (const float*)d_in[19];
  const float* bi   = (const float*)d_in[20];
  const float* Gw1  = (const float*)d_in[21];
  const float* Gb1  = (const float*)d_in[22];
  const float* Gw2  = (const float*)d_in[23];
  const float* Gb2  = (const float*)d_in[24];
  const float* Ws   = (const float*)d_in[25];
  const float* bs   = (const float*)d_in[26];
  float* out = (float*)d_out;

  // ---- workspace layout (floats) ----
  float* ws    = (float*)d_ws;
  float* dinv  = ws;                            // N
  float* bufA  = dinv + NN;                     // N*H  (h1 / xin / h3 / h4)
  float* bufB  = bufA + (size_t)NN * H_;        // N*H  (x1, later gate hidden)
  float* bufC  = bufB + (size_t)NN * H_;        // N*H  (hg, later x3)
  float* bufD  = bufC + (size_t)NN * H_;        // N*H  (x2)
  float* as_   = bufD + (size_t)NN * H_;        // N*4
  float* ad_   = as_  + (size_t)NN * NHEADS_;
  float* emax  = ad_  + (size_t)NN * NHEADS_;
  float* esum  = emax + (size_t)NN * NHEADS_;
  float* stats = esum + (size_t)NN * NHEADS_;   // 2*H
  float* kAB   = stats + 2 * H_;                // 2*H
  float* xout4 = kAB + 2 * H_;                  // N*12
  float* gout  = xout4 + (size_t)NN * DOUT_;    // N*12
  float* exbuf = gout  + (size_t)NN * DOUT_;    // E*4
  // transposed (zero-padded) weights
  float* W1t  = exbuf + (size_t)EE * NHEADS_;   // 128*24
  float* Wit  = W1t  + 128 * 24;                // 128*24
  float* Wgt  = Wit  + 128 * 24;                // 128*128
  float* W3t  = Wgt  + 128 * 128;               // 128*128
  float* W4t  = W3t  + 128 * 128;               // 16*128
  float* Gw1t = W4t  + 16 * 128;                // 64*128
  float* Gw2t = Gw1t + 64 * 128;                // 16*64

  auto blocks = [](long long n) { return (unsigned)((n + 255) / 256); };
  auto transpose = [&](const float* B, float* Bt, int K, int P, int Pp) {
    k_transpose<<<blocks((long long)Pp * K), 256, 0, stream>>>(B, Bt, K, P, Pp);
  };

  // one-shot weight transposes (cheap, deterministic every call)
  transpose(W1,  W1t,  DIN_, H_,    H_);
  transpose(Wi,  Wit,  DIN_, H_,    H_);
  transpose(Wg,  Wgt,  H_,   H_,    H_);
  transpose(W3,  W3t,  H_,   H_,    H_);
  transpose(W4,  W4t,  H_,   DOUT_, 16);
  transpose(Gw1, Gw1t, H_,   64,    64);
  transpose(Gw2, Gw2t, 64,   DOUT_, 16);

  // degrees -> dinv
  k_fill1<<<blocks(NN), 256, 0, stream>>>(dinv, NN);
  k_deg_edges<<<blocks(EE), 256, 0, stream>>>(dinv, ei, ew, EE);
  k_dinv<<<blocks(NN), 256, 0, stream>>>(dinv, NN);

  // ---- layer 1: x1 = relu(bn(gcn(x, W1) + b1)) ----
  k_gemm_wmma<DIN_, H_><<<6250, 256, 0, stream>>>(x, W1t, nullptr, bufA, 0);      // h1
  k_gcn_self128v<<<blocks((long long)NN * 32), 256, 0, stream>>>(bufA, dinv, bufB);
  k_gcn_edges128v<<<blocks((long long)EE * 32), 256, 0, stream>>>(bufA, dinv, ew, ei, bufB);
  hipMemsetAsync(stats, 0, 2 * H_ * sizeof(float), stream);
  k_bn_stats<<<blocks(NN), H_, 0, stream>>>(bufB, b1, stats);
  k_bn_finalize<<<1, H_, 0, stream>>>(stats, g1, be1, kAB);
  k_bn_apply<<<blocks((long long)NN * 32), 256, 0, stream>>>(bufB, b1, kAB, nullptr, bufB); // x1

  // ---- input projection + GAT ----
  k_gemm_wmma<DIN_, H_><<<6250, 256, 0, stream>>>(x, Wit, bi, bufA, 0);           // xin
  k_gemm_wmma<H_, H_><<<6250, 256, 0, stream>>>(bufB, Wgt, nullptr, bufC, 0);     // hg
  k_att_scores<<<blocks((long long)NN * NHEADS_), 256, 0, stream>>>(bufC, asv, adv, as_, ad_);
  k_att_maxinit<<<blocks((long long)NN * NHEADS_), 256, 0, stream>>>(as_, ad_, emax);
  k_att_max<<<blocks((long long)EE * NHEADS_), 256, 0, stream>>>(ei, as_, ad_, emax);
  k_att_suminit<<<blocks((long long)NN * NHEADS_), 256, 0, stream>>>(as_, ad_, emax, esum);
  k_att_sum<<<blocks((long long)EE * NHEADS_), 256, 0, stream>>>(ei, as_, ad_, emax, esum, exbuf);
  k_att_aggselfv<<<blocks((long long)NN * 32), 256, 0, stream>>>(bufC, as_, ad_, emax, esum, bufD);
  k_att_aggedgesv<<<blocks((long long)EE * 32), 256, 0, stream>>>(ei, bufC, exbuf, esum, bufD);
  hipMemsetAsync(stats, 0, 2 * H_ * sizeof(float), stream);
  k_bn_stats<<<blocks(NN), H_, 0, stream>>>(bufD, bg, stats);
  k_bn_finalize<<<1, H_, 0, stream>>>(stats, g2, be2, kAB);
  k_bn_apply<<<blocks((long long)NN * 32), 256, 0, stream>>>(bufD, bg, kAB, bufA, bufD); // x2

  // ---- layer 3: x3 = relu(bn(gcn(x2, W3) + b3)) + x1 ----
  k_gemm_wmma<H_, H_><<<6250, 256, 0, stream>>>(bufD, W3t, nullptr, bufA, 0);     // h3
  k_gcn_self128v<<<blocks((long long)NN * 32), 256, 0, stream>>>(bufA, dinv, bufC);
  k_gcn_edges128v<<<blocks((long long)EE * 32), 256, 0, stream>>>(bufA, dinv, ew, ei, bufC);
  hipMemsetAsync(stats, 0, 2 * H_ * sizeof(float), stream);
  k_bn_stats<<<blocks(NN), H_, 0, stream>>>(bufC, b3, stats);
  k_bn_finalize<<<1, H_, 0, stream>>>(stats, g3, be3, kAB);
  k_bn_apply<<<blocks((long long)NN * 32), 256, 0, stream>>>(bufC, b3, kAB, bufB, bufC); // x3

  // ---- output head: x_out = gcn(x3, W4) + b4 ----
  k_gemm_wmma<H_, DOUT_><<<782, 256, 0, stream>>>(bufC, W4t, nullptr, bufA, 0);   // h4 (N x 12)
  k_gcn_self<DOUT_><<<blocks((long long)NN * DOUT_), 256, 0, stream>>>(bufA, dinv, xout4);
  k_gcn_edges<DOUT_><<<blocks((long long)EE * DOUT_), 256, 0, stream>>>(bufA, dinv, ew, ei, xout4);

  // ---- gating MLP (sigmoid fused into final) ----
  k_gemm_wmma<H_, 64><<<3125, 256, 0, stream>>>(bufC, Gw1t, Gb1, bufB, 1);        // relu hidden
  k_gemm_wmma<64, DOUT_><<<782, 256, 0, stream>>>(bufB, Gw2t, Gb2, gout, 0);      // gate logits

  k_final<<<blocks((long long)NN * DOUT_), 256, 0, stream>>>(x, xout4, gout, Ws, bs, b4, out);
  (void)in_sizes; (void)n_in; (void)out_size; (void)ws_size;
}